// GeometricEncoder_70368744177849
// MI455X (gfx1250) — compile-verified
//
#include <hip/hip_runtime.h>
#include <hip/hip_bf16.h>
#include <math.h>

typedef __attribute__((ext_vector_type(2))) float v2f;
typedef __attribute__((ext_vector_type(8))) float v8f;

#define NF 256
#define XCOLS 33
#define KPAD 36
#define TM 128
#define TN 64
#define TK 32
#define APAD 1   // v2f pad -> row stride 17 v2f (34 banks, 2*odd): conflict-free
#define BPAD 16  // v2f pad -> row stride 80 v2f (160 % 64 == 32): conflict-free

// ---------------- ordered-int float max encoding ----------------
__device__ __forceinline__ unsigned int encodeOrd(float f) {
    unsigned int u = __float_as_uint(f);
    return (u & 0x80000000u) ? ~u : (u | 0x80000000u);
}

// ---------------- reduction: rm = max(X[:,27]) / 6.283 ----------------
__global__ void k_reduce_rm(const float* __restrict__ X, int n, float* __restrict__ rm) {
    __shared__ float red[256];
    float m = -INFINITY;
    for (int i = threadIdx.x; i < n; i += blockDim.x)
        m = fmaxf(m, X[(long)i * XCOLS + 27]);
    red[threadIdx.x] = m; __syncthreads();
    for (int s = 128; s > 0; s >>= 1) {
        if (threadIdx.x < (unsigned)s) red[threadIdx.x] = fmaxf(red[threadIdx.x], red[threadIdx.x + s]);
        __syncthreads();
    }
    if (threadIdx.x == 0) rm[0] = red[0] / 6.283f;
}

// ---------------- preprocess: build x_in (n x 36, zero padded) ----------------
__global__ void k_preprocess(const float* __restrict__ X, int n, const float* __restrict__ rm,
                             float* __restrict__ xin) {
    int i = blockIdx.x * blockDim.x + threadIdx.x;
    if (i >= n * KPAD) return;
    int r = i / KPAD, c = i % KPAD;
    const float* xr = X + (long)r * XCOLS;
    float v;
    if (c < 27)       v = xr[c];
    else if (c == 27) v = sinf(xr[27] / rm[0]);
    else if (c == 28) v = cosf(xr[27] / rm[0]);
    else if (c <= 32) v = xr[c - 1];          // x0 cols 28..31
    else              v = 0.0f;               // pad
    xin[i] = v;
}

// ---------------- pad 33x256 weight to 36x256 ----------------
__global__ void k_pad_w(const float* __restrict__ W, float* __restrict__ Wp) {
    int i = blockIdx.x * blockDim.x + threadIdx.x;
    if (i >= KPAD * NF) return;
    int r = i / NF, c = i % NF;
    Wp[i] = (r < XCOLS) ? W[r * NF + c] : 0.0f;
}

// ---------------- WMMA f32 GEMM: C[MxNF] = A[MxK] * W[KxNF] (+bias0+bias1) ----------------
// LDS holds k-pair-interleaved tiles so every fragment is a single ds_load_b64:
//   sAp[row][q] = { A[row][2q], A[row][2q+1] }   (A frag: q = kk/2 + (lane>>4))
//   sBp[q][n]   = { W[2q][n],  W[2q+1][n]  }     (B frag: same q, n = t*16 + lane&15)
__global__ __launch_bounds__(256)
void k_gemm_wmma(const float* __restrict__ A, int lda, int K,
                 const float* __restrict__ W,
                 const float* __restrict__ bias0, const float* __restrict__ bias1,
                 float* __restrict__ C, int M) {
    __shared__ v2f sAp[TM][TK / 2 + APAD];
    __shared__ v2f sBp[TK / 2][TN + BPAD];

    const int tid  = threadIdx.x;
    const int lane = tid & 31;
    const int wave = tid >> 5;
    const int m0   = blockIdx.x * TM;
    const int n0   = blockIdx.y * TN;

    v8f acc[4] = {};

    const int arow = wave * 16 + (lane & 15);
    const int kq   = lane >> 4;          // which row of the k-pair quartet
    const int ncol = lane & 15;

    for (int kb = 0; kb < K; kb += TK) {
        const int kc = (K - kb) < TK ? (K - kb) : TK;   // multiple of 4
        // stage A tile as k-pairs (aligned float2 global loads, coalesced per row)
        for (int i = tid; i < TM * (TK / 2); i += 256) {
            int r = i / (TK / 2), c2 = i % (TK / 2);
            int gr = m0 + r;
            v2f v = {};
            if (2 * c2 < kc && gr < M) {
                const float* ap = A + (long)gr * lda + kb + 2 * c2;
                v.x = ap[0]; v.y = ap[1];
            }
            sAp[r][c2] = v;
        }
        // stage B tile as k-pairs
        for (int i = tid; i < (TK / 2) * TN; i += 256) {
            int r2 = i / TN, c = i % TN;
            v2f v = {};
            if (2 * r2 < kc) {
                const float* wp = W + (long)(kb + 2 * r2) * NF + n0 + c;
                v.x = wp[0]; v.y = wp[NF];
            }
            sBp[r2][c] = v;
        }
        __syncthreads();

        // software-pipelined fragment loads: prefetch next k-step behind 4 WMMAs
        v2f aF  = sAp[arow][kq];
        v2f bF0 = sBp[kq][0 * 16 + ncol];
        v2f bF1 = sBp[kq][1 * 16 + ncol];
        v2f bF2 = sBp[kq][2 * 16 + ncol];
        v2f bF3 = sBp[kq][3 * 16 + ncol];
#pragma unroll
        for (int kk = 0; kk < TK; kk += 4) {
            v2f aN = aF, bN0 = bF0, bN1 = bF1, bN2 = bF2, bN3 = bF3;
            if (kk + 4 < TK) {
                int q = ((kk + 4) >> 1) + kq;
                aN  = sAp[arow][q];
                bN0 = sBp[q][0 * 16 + ncol];
                bN1 = sBp[q][1 * 16 + ncol];
                bN2 = sBp[q][2 * 16 + ncol];
                bN3 = sBp[q][3 * 16 + ncol];
            }
            acc[0] = __builtin_amdgcn_wmma_f32_16x16x4_f32(false, aF, false, bF0, (short)0, acc[0], false, false);
            acc[1] = __builtin_amdgcn_wmma_f32_16x16x4_f32(false, aF, false, bF1, (short)0, acc[1], false, false);
            acc[2] = __builtin_amdgcn_wmma_f32_16x16x4_f32(false, aF, false, bF2, (short)0, acc[2], false, false);
            acc[3] = __builtin_amdgcn_wmma_f32_16x16x4_f32(false, aF, false, bF3, (short)0, acc[3], false, false);
            aF = aN; bF0 = bN0; bF1 = bN1; bF2 = bN2; bF3 = bN3;
        }
        __syncthreads();
    }

    const int rbase = m0 + wave * 16 + ((lane >> 4) << 3);
#pragma unroll
    for (int t = 0; t < 4; ++t) {
        const int col = n0 + t * 16 + ncol;
        float badd = 0.0f;
        if (bias0) badd += bias0[col];
        if (bias1) badd += bias1[col];
#pragma unroll
        for (int r = 0; r < 8; ++r) {
            int gm = rbase + r;
            if (gm < M) C[(long)gm * NF + col] = acc[t][r] + badd;
        }
    }
}

// ---------------- per-node per-head attention scores ----------------
__global__ void k_head_scores(const float* __restrict__ h, const float* __restrict__ asrc,
                              const float* __restrict__ adst, int n,
                              float* __restrict__ ssrc, float* __restrict__ sdst) {
    int i = blockIdx.x * blockDim.x + threadIdx.x;
    if (i >= n * 8) return;
    int node = i >> 3, hh = i & 7;
    const float* hp = h + (long)node * NF + hh * 32;
    const float* as = asrc + hh * 32;
    const float* ad = adst + hh * 32;
    float s1 = 0.f, s2 = 0.f;
#pragma unroll
    for (int d = 0; d < 32; ++d) { float v = hp[d]; s1 += v * as[d]; s2 += v * ad[d]; }
    ssrc[i] = s1; sdst[i] = s2;
}

// ---------------- init segment max / denom ----------------
__global__ void k_init_attn(unsigned int* __restrict__ mmax, float* __restrict__ denom, int n8) {
    int i = blockIdx.x * blockDim.x + threadIdx.x;
    if (i >= n8) return;
    mmax[i]  = 0x007FFFFFu;  // encodeOrd(-inf)
    denom[i] = 0.0f;
}

// ---------------- edge pass 1: logits + segment max ----------------
__global__ void k_edge_logits_max(const int* __restrict__ E, const float* __restrict__ Ea,
                                  int ne, int n,
                                  const float* __restrict__ ssrc, const float* __restrict__ sdst,
                                  float* __restrict__ elog, unsigned int* __restrict__ mmax,
                                  int hasEa) {
    int e = blockIdx.x * blockDim.x + threadIdx.x;
    int M = ne + n;
    if (e >= M) return;
    int s, d; float ea = 0.0f;
    if (e < ne) { s = E[2 * e]; d = E[2 * e + 1]; if (hasEa) ea = Ea[e]; }
    else        { s = d = e - ne; }
#pragma unroll
    for (int hh = 0; hh < 8; ++hh) {
        float l = ssrc[s * 8 + hh] + sdst[d * 8 + hh] + ea;
        l = (l > 0.0f) ? l : 0.2f * l;          // leaky_relu(0.2)
        elog[(long)e * 8 + hh] = l;
        atomicMax(&mmax[d * 8 + hh], encodeOrd(l));
    }
}

// ---------------- decode ordered-int max back to float (in place) ----------------
__global__ void k_decode_max(unsigned int* __restrict__ mmax, int n8) {
    int i = blockIdx.x * blockDim.x + threadIdx.x;
    if (i >= n8) return;
    unsigned int u = mmax[i];
    float f = (u & 0x80000000u) ? __uint_as_float(u & 0x7FFFFFFFu) : __uint_as_float(~u);
    ((float*)mmax)[i] = f;
}

// ---------------- edge pass 2: exp + denom ----------------
__global__ void k_edge_exp_sum(const int* __restrict__ E, int ne, int n,
                               const float* __restrict__ m,
                               float* __restrict__ elog, float* __restrict__ denom) {
    int i = blockIdx.x * blockDim.x + threadIdx.x;
    int M8 = (ne + n) * 8;
    if (i >= M8) return;
    int e = i >> 3, hh = i & 7;
    int d = (e < ne) ? E[2 * e + 1] : (e - ne);
    float v = expf(elog[i] - m[d * 8 + hh]);
    elog[i] = v;
    atomicAdd(&denom[d * 8 + hh], v);
}

// ---------------- denom -> 1/denom ----------------
__global__ void k_recip(float* __restrict__ d, int n8) {
    int i = blockIdx.x * blockDim.x + threadIdx.x;
    if (i < n8) d[i] = 1.0f / d[i];
}

// ---------------- edge pass 3: scatter-add alpha * h[src] into acc[dst] ----------------
__global__ __launch_bounds__(256)
void k_edge_scatter(const int* __restrict__ E, int ne, int n,
                    const float* __restrict__ elog, const float* __restrict__ rden,
                    const float* __restrict__ h, float* __restrict__ acc) {
    int e = blockIdx.x;
    int c = threadIdx.x;
    int s, d;
    if (e < ne) { s = E[2 * e]; d = E[2 * e + 1]; }
    else        { s = d = e - ne; }
    int hh = c >> 5;
    float alpha = elog[(long)e * 8 + hh] * rden[d * 8 + hh];
    atomicAdd(&acc[(long)d * NF + c], alpha * h[(long)s * NF + c]);
}

// ---------------- ELU ----------------
__global__ void k_elu(const float* __restrict__ acc, float* __restrict__ out, long total) {
    long i = (long)blockIdx.x * blockDim.x + threadIdx.x;
    if (i >= total) return;
    float v = acc[i];
    out[i] = (v > 0.0f) ? v : (expf(v) - 1.0f);
}

// ---------------- masked per-column max & sum ----------------
__global__ void k_masked_stats(const float* __restrict__ x, const float* __restrict__ X,
                               int n, int maskcol,
                               float* __restrict__ mxOut, float* __restrict__ smOut) {
    int col = blockIdx.x;
    int tid = threadIdx.x;
    float mx = -INFINITY, sm = 0.0f;
    for (int r = tid; r < n; r += blockDim.x) {
        if (X[(long)r * XCOLS + maskcol] == 1.0f) {
            float v = x[(long)r * NF + col];
            mx = fmaxf(mx, v); sm += v;
        }
    }
    __shared__ float rmx[256], rsm[256];
    rmx[tid] = mx; rsm[tid] = sm; __syncthreads();
    for (int s = 128; s > 0; s >>= 1) {
        if (tid < s) { rmx[tid] = fmaxf(rmx[tid], rmx[tid + s]); rsm[tid] += rsm[tid + s]; }
        __syncthreads();
    }
    if (tid == 0) { mxOut[col] = rmx[0]; smOut[col] = rsm[0]; }
}

// ---------------- mask counts: [iface(col32), cont(col24)] ----------------
__global__ void k_mask_counts(const float* __restrict__ X, int n, float* __restrict__ cnt) {
    __shared__ float c1[256], c2[256];
    int tid = threadIdx.x;
    float a = 0.f, b = 0.f;
    for (int r = tid; r < n; r += blockDim.x) {
        const float* xr = X + (long)r * XCOLS;
        if (xr[32] == 1.0f) a += 1.0f;
        if (xr[24] == 1.0f) b += 1.0f;
    }
    c1[tid] = a; c2[tid] = b; __syncthreads();
    for (int s = 128; s > 0; s >>= 1) {
        if (tid < s) { c1[tid] += c1[tid + s]; c2[tid] += c2[tid + s]; }
        __syncthreads();
    }
    if (tid == 0) { cnt[0] = c1[0]; cnt[1] = c2[0]; }
}

// ---------------- final assembly of 18x256 output ----------------
// stats layout per graph (8 arrays of 256): mx4_if, s4_if, mx4_c, s4_c, mx3_if, s3_if, mx3_c, s3_c
__global__ void k_assemble(const float* __restrict__ g0, const float* __restrict__ g1,
                           const float* __restrict__ cnt0, const float* __restrict__ cnt1,
                           float* __restrict__ out) {
    int c = blockIdx.x * blockDim.x + threadIdx.x;
    if (c >= NF) return;
    float mx4   = g0[0*NF+c], s4    = g0[1*NF+c], mx4c  = g0[2*NF+c], s4c   = g0[3*NF+c];
    float mx3   = g0[4*NF+c], s3    = g0[5*NF+c], mx3c  = g0[6*NF+c], s3c   = g0[7*NF+c];
    float mx4m  = g1[0*NF+c], s4m   = g1[1*NF+c], mx4mc = g1[2*NF+c], s4mc  = g1[3*NF+c];
    float mx3m  = g1[4*NF+c], s3m   = g1[5*NF+c], mx3mc = g1[6*NF+c], s3mc  = g1[7*NF+c];
    float mn3   = s3  / cnt0[0], mn3m  = s3m  / cnt1[0];
    float mn3c  = s3c / cnt0[1], mn3mc = s3mc / cnt1[1];
    out[ 0*NF+c] = mx4;   out[ 1*NF+c] = mx4m;  out[ 2*NF+c] = s4;    out[ 3*NF+c] = s4m;
    out[ 4*NF+c] = mx4c;  out[ 5*NF+c] = mx4mc; out[ 6*NF+c] = s4c;   out[ 7*NF+c] = s4mc;
    out[ 8*NF+c] = mx3;   out[ 9*NF+c] = mx3m;  out[10*NF+c] = mn3;   out[11*NF+c] = mn3m;
    out[12*NF+c] = mx3c;  out[13*NF+c] = mx3mc; out[14*NF+c] = mn3c;  out[15*NF+c] = mn3mc;
    out[16*NF+c] = mx4 - mx4m;                  out[17*NF+c] = s4 - s4m;
}

// ============================ host orchestration ============================
extern "C" void kernel_launch(void* const* d_in, const int* in_sizes, int n_in,
                              void* d_out, int out_size, void* d_ws, size_t ws_size,
                              hipStream_t stream) {
    const int n  = in_sizes[0] / XCOLS;   // 20000
    const int ne = in_sizes[2];           // 400000
    const int M  = ne + n;                // edges incl. self-loops
    const int n8 = n * 8;

    const float* Xg[2]  = { (const float*)d_in[0], (const float*)d_in[3] };
    const int*   Eg[2]  = { (const int*)  d_in[1], (const int*)  d_in[4] };
    const float* Eag[2] = { (const float*)d_in[2], (const float*)d_in[5] };

    const float *Wc[4], *Asr[4], *Ads[4], *Bc[4], *Wl[4], *Bl[4];
    for (int i = 0; i < 4; ++i) {
        int b = 6 + 6 * i;
        Wc[i]  = (const float*)d_in[b + 0];
        Asr[i] = (const float*)d_in[b + 1];
        Ads[i] = (const float*)d_in[b + 2];
        Bc[i]  = (const float*)d_in[b + 3];
        Wl[i]  = (const float*)d_in[b + 4];
        Bl[i]  = (const float*)d_in[b + 5];
    }

    // ---- workspace layout (floats) ----
    float* w = (float*)d_ws;
    size_t o = 0;
    auto take = [&](size_t cnt) { float* p = w + o; o += cnt; return p; };
    float* rm    = take(1);
    float* xin   = take((size_t)n * KPAD);
    float* WcP   = take((size_t)KPAD * NF);
    float* WlP   = take((size_t)KPAD * NF);
    float* hbuf  = take((size_t)n * NF);
    float* acc   = take((size_t)n * NF);
    float* xA    = take((size_t)n * NF);
    float* xB    = take((size_t)n * NF);
    float* x3    = take((size_t)n * NF);
    float* x4    = take((size_t)n * NF);
    float* ssrc  = take(n8);
    float* sdst  = take(n8);
    unsigned int* mmax = (unsigned int*)take(n8);
    float* denom = take(n8);
    float* elog  = take((size_t)M * 8);
    float* stats[2] = { take(8 * NF), take(8 * NF) };
    float* cnts[2]  = { take(2), take(2) };
    (void)ws_size; (void)n_in; (void)out_size;

    // pad layer-1 weights once (shared by both graphs)
    {
        int tot = KPAD * NF;
        k_pad_w<<<(tot + 255) / 256, 256, 0, stream>>>(Wc[0], WcP);
        k_pad_w<<<(tot + 255) / 256, 256, 0, stream>>>(Wl[0], WlP);
    }

    dim3 gemmGrid((n + TM - 1) / TM, NF / TN);

    for (int g = 0; g < 2; ++g) {
        const float* X  = Xg[g];
        const int*   E  = Eg[g];
        const float* Ea = Eag[g];

        k_reduce_rm<<<1, 256, 0, stream>>>(X, n, rm);
        k_preprocess<<<((n * KPAD) + 255) / 256, 256, 0, stream>>>(X, n, rm, xin);

        const float* xcur = xin;
        for (int L = 0; L < 4; ++L) {
            const int   K   = (L == 0) ? KPAD : NF;
            const float* Wa = (L == 0) ? WcP : Wc[L];
            const float* Wb = (L == 0) ? WlP : Wl[L];
            float* xnext = (L == 0) ? xA : (L == 1) ? xB : (L == 2) ? x3 : x4;

            // h = x @ Wc ; acc = x @ Wl + bl + bc
            k_gemm_wmma<<<gemmGrid, 256, 0, stream>>>(xcur, K, K, Wa, nullptr, nullptr, hbuf, n);
            k_gemm_wmma<<<gemmGrid, 256, 0, stream>>>(xcur, K, K, Wb, Bl[L], Bc[L], acc, n);

            k_head_scores<<<(n8 + 255) / 256, 256, 0, stream>>>(hbuf, Asr[L], Ads[L], n, ssrc, sdst);
            k_init_attn<<<(n8 + 255) / 256, 256, 0, stream>>>(mmax, denom, n8);
            k_edge_logits_max<<<(M + 255) / 256, 256, 0, stream>>>(E, Ea, ne, n, ssrc, sdst,
                                                                   elog, mmax, (L == 0) ? 1 : 0);
            k_decode_max<<<(n8 + 255) / 256, 256, 0, stream>>>(mmax, n8);
            k_edge_exp_sum<<<((M * 8) + 255) / 256, 256, 0, stream>>>(E, ne, n, (const float*)mmax,
                                                                      elog, denom);
            k_recip<<<(n8 + 255) / 256, 256, 0, stream>>>(denom, n8);
            k_edge_scatter<<<M, 256, 0, stream>>>(E, ne, n, elog, denom, hbuf, acc);
            k_elu<<<n, 256, 0, stream>>>(acc, xnext, (long)n * NF);

            xcur = xnext;
        }

        k_mask_counts<<<1, 256, 0, stream>>>(X, n, cnts[g]);
        float* st = stats[g];
        k_masked_stats<<<NF, 256, 0, stream>>>(x4, X, n, 32, st + 0 * NF, st + 1 * NF);
        k_masked_stats<<<NF, 256, 0, stream>>>(x4, X, n, 24, st + 2 * NF, st + 3 * NF);
        k_masked_stats<<<NF, 256, 0, stream>>>(x3, X, n, 32, st + 4 * NF, st + 5 * NF);
        k_masked_stats<<<NF, 256, 0, stream>>>(x3, X, n, 24, st + 6 * NF, st + 7 * NF);
    }

    k_assemble<<<1, 256, 0, stream>>>(stats[0], stats[1], cnts[0], cnts[1], (float*)d_out);
}